// SelfContactOptiLoss_77240691851886
// MI455X (gfx1250) — compile-verified
//
#include <hip/hip_runtime.h>

// ---------------------------------------------------------------------------
// SelfContactOptiLoss for MI455X (gfx1250, wave32).
// Dominant cost: one streaming pass over the 439MB geodist matrix (min over
// 500 gathered columns per row) -> coalesced wave-per-row, aligned b128
// non-temporal loads (geodist > 192MB L2), column bitmask in LDS so unneeded
// cache lines are never requested. All scalar-sum reductions use a
// V_WMMA_F32_16X16X4_F32-based wave32 reduction (no shuffle needed: both
// half-sums land replicated in lane groups 0-15 / 16-31, lanes 0 and 16
// each contribute their half via atomicAdd).
// ---------------------------------------------------------------------------

#define NVERT 10475
#define NDSC  2048
#define NFACE 20908
#define HAC   778
#define NCON  500
#define COLW  ((NVERT + 31) / 32)

typedef __attribute__((ext_vector_type(2))) float v2f;
typedef __attribute__((ext_vector_type(4))) float v4f;
typedef __attribute__((ext_vector_type(8))) float v8f;

// Wave32 sum reduction via WMMA f32 16x16x4.
// A layout (32-bit A 16x4): lane m (0-15): VGPR0->K=0; lane 16+m: VGPR0->K=2.
// With a = {x, 0} and B = all-ones: D[m,n] = x(m) + x(m+16) for every n.
// D layout: lane L(0-15) holds column N=L rows M=0..7; lane 16+L holds M=8..15.
// Lane-local sum of the 8 D VGPRs => lanes 0-15 all hold S_low, lanes 16-31
// all hold S_high, with S_low + S_high = full wave sum. Caller: lanes with
// (lane & 15) == 0 atomicAdd their half. Requires all 32 lanes active.
__device__ __forceinline__ float wave_sum_half(float x) {
  v2f a; a[0] = x;    a[1] = 0.0f;
  v2f b; b[0] = 1.0f; b[1] = 1.0f;
  v8f c = {};
  v8f d = __builtin_amdgcn_wmma_f32_16x16x4_f32(
      /*neg_a=*/false, a, /*neg_b=*/false, b,
      /*c_mod=*/(short)0, c, /*reuse_a=*/false, /*reuse_b=*/false);
  return d[0] + d[1] + d[2] + d[3] + d[4] + d[5] + d[6] + d[7];
}

// Accumulator slots (float):
//  0: contact sum     1: contact cnt    2: inside sum     3: inside cnt
//  4: l_out sum       5: l_out cnt      6: r_out sum      7: r_out cnt
//  8: l_in sum        9: l_in cnt      10: r_in sum      11: r_in cnt
// 12: angle sum      13: angle cnt     14: pose sum      15: handpose sum
// 16: outside sum    17-19: vn sums (for nbar)

__global__ void k_zero(unsigned* __restrict__ w, int nwords) {
  int i = blockIdx.x * blockDim.x + threadIdx.x;
  if (i < nwords) w[i] = 0u;
}

__global__ void k_scatter(const int* __restrict__ ds,
                          const unsigned char* __restrict__ exterior,
                          const int* __restrict__ contact,
                          unsigned* __restrict__ inside,
                          unsigned* __restrict__ colmask) {
  int i = blockIdx.x * blockDim.x + threadIdx.x;
  if (i < NDSC) inside[ds[i]] = exterior[i] ? 0u : 1u;
  if (i < NCON) {
    int c = contact[i];
    atomicOr(&colmask[c >> 5], 1u << (c & 31));
  }
}

// One wave per row. Rows are DWORD-aligned only (NVERT odd): scalar prologue
// of (row & 3) columns reaches 16B alignment, then a b128-per-lane NT body
// (512B per wave iteration), then a scalar tail. Loads are predicated on the
// contact-column bitmask so fully-unneeded 128B lines are never fetched.
__global__ void k_gather_min(const float* __restrict__ geodist,
                             const unsigned* __restrict__ colmask,
                             float* __restrict__ gd) {
  __shared__ unsigned smask[COLW];
  for (int i = threadIdx.x; i < COLW; i += blockDim.x) smask[i] = colmask[i];
  __syncthreads();
  const int lane = threadIdx.x & 31;
  const int row  = blockIdx.x * (blockDim.x >> 5) + (threadIdx.x >> 5);
  if (row >= NVERT) return;
  const float* rp = geodist + (size_t)row * NVERT;
  float m = 3.0e38f;

  const int p = row & 3;  // columns to 16B alignment (row offset = 12*row mod 16)
  if (lane < p) {
    int c = lane;
    if ((smask[c >> 5] >> (c & 31)) & 1u)
      m = fminf(m, __builtin_nontemporal_load(rp + c));
  }
  const int body_end = p + ((NVERT - p) & ~127);
  for (int base = p; base < body_end; base += 128) {
    int c = base + 4 * lane;
    unsigned b0 = (smask[(c + 0) >> 5] >> ((c + 0) & 31)) & 1u;
    unsigned b1 = (smask[(c + 1) >> 5] >> ((c + 1) & 31)) & 1u;
    unsigned b2 = (smask[(c + 2) >> 5] >> ((c + 2) & 31)) & 1u;
    unsigned b3 = (smask[(c + 3) >> 5] >> ((c + 3) & 31)) & 1u;
    if (b0 | b1 | b2 | b3) {
      v4f v = __builtin_nontemporal_load((const v4f*)(rp + c));
      if (b0) m = fminf(m, v[0]);
      if (b1) m = fminf(m, v[1]);
      if (b2) m = fminf(m, v[2]);
      if (b3) m = fminf(m, v[3]);
    }
  }
  for (int c = body_end + lane; c < NVERT; c += 32) {
    if ((smask[c >> 5] >> (c & 31)) & 1u)
      m = fminf(m, __builtin_nontemporal_load(rp + c));
  }
  #pragma unroll
  for (int o = 16; o > 0; o >>= 1) m = fminf(m, __shfl_xor(m, o, 32));
  if (lane == 0) gd[row] = m;
}

__global__ void k_face_normals(const float* __restrict__ verts,
                               const int* __restrict__ faces,
                               float* __restrict__ vn) {
  int f = blockIdx.x * blockDim.x + threadIdx.x;
  if (f >= NFACE) return;
  int i0 = faces[3 * f + 0], i1 = faces[3 * f + 1], i2 = faces[3 * f + 2];
  float ax = verts[3 * i0], ay = verts[3 * i0 + 1], az = verts[3 * i0 + 2];
  float e1x = verts[3 * i1]     - ax, e1y = verts[3 * i1 + 1] - ay, e1z = verts[3 * i1 + 2] - az;
  float e2x = verts[3 * i2]     - ax, e2y = verts[3 * i2 + 1] - ay, e2z = verts[3 * i2 + 2] - az;
  float nx = e1y * e2z - e1z * e2y;
  float ny = e1z * e2x - e1x * e2z;
  float nz = e1x * e2y - e1y * e2x;
  atomicAdd(&vn[3 * i0 + 0], nx); atomicAdd(&vn[3 * i0 + 1], ny); atomicAdd(&vn[3 * i0 + 2], nz);
  atomicAdd(&vn[3 * i1 + 0], nx); atomicAdd(&vn[3 * i1 + 1], ny); atomicAdd(&vn[3 * i1 + 2], nz);
  atomicAdd(&vn[3 * i2 + 0], nx); atomicAdd(&vn[3 * i2 + 1], ny); atomicAdd(&vn[3 * i2 + 2], nz);
}

__global__ void k_normalize_vn(float* __restrict__ vn, float* __restrict__ acc) {
  int v = blockIdx.x * blockDim.x + threadIdx.x;
  float x = 0.f, y = 0.f, z = 0.f;
  if (v < NVERT) {
    x = vn[3 * v]; y = vn[3 * v + 1]; z = vn[3 * v + 2];
    float inv = 1.0f / (sqrtf(x * x + y * y + z * z) + 1e-12f);
    x *= inv; y *= inv; z *= inv;
    vn[3 * v] = x; vn[3 * v + 1] = y; vn[3 * v + 2] = z;
  }
  float sx = wave_sum_half(x), sy = wave_sum_half(y), sz = wave_sum_half(z);
  if ((threadIdx.x & 15) == 0) {  // lanes 0 and 16: each adds its half-sum
    atomicAdd(&acc[17], sx); atomicAdd(&acc[18], sy); atomicAdd(&acc[19], sz);
  }
}

__global__ void k_vertex_terms(const float* __restrict__ verts,
                               const float* __restrict__ iverts,
                               const float* __restrict__ v2v_min,
                               const int* __restrict__ v2v_idx,
                               const float* __restrict__ gd,
                               const unsigned* __restrict__ inside,
                               const float* __restrict__ vn,
                               float* __restrict__ acc) {
  const float inv_nv = 1.0f / (float)NVERT;
  const float nbx = acc[17] * inv_nv, nby = acc[18] * inv_nv, nbz = acc[19] * inv_nv;
  int v = blockIdx.x * blockDim.x + threadIdx.x;
  float o = 0.f, is = 0.f, ic = 0.f, as = 0.f, ac = 0.f;
  if (v < NVERT) {
    float dx = iverts[3 * v] - verts[3 * v];
    float dy = iverts[3 * v + 1] - verts[3 * v + 1];
    float dz = iverts[3 * v + 2] - verts[3 * v + 2];
    float v2v0 = sqrtf(dx * dx + dy * dy + dz * dz);
    float g = gd[v];
    o = v2v0 * (4.0f * g * g);                 // v2v0 * (2*gd)^2
    float vm = v2v_min[v];
    if (inside[v]) { is = 0.07f * tanhf(vm * (1.0f / 0.06f)); ic = 1.0f; }  // B1,B2
    if (vm < 0.01f) {
      int j = v2v_idx[v];
      as = 1.0f + vn[3 * j] * nbx + vn[3 * j + 1] * nby + vn[3 * j + 2] * nbz;
      ac = 1.0f;
    }
  }
  float so = wave_sum_half(o), s2 = wave_sum_half(is), s3 = wave_sum_half(ic);
  float s12 = wave_sum_half(as), s13 = wave_sum_half(ac);
  if ((threadIdx.x & 15) == 0) {
    atomicAdd(&acc[16], so);
    atomicAdd(&acc[2], s2);  atomicAdd(&acc[3], s3);
    atomicAdd(&acc[12], s12); atomicAdd(&acc[13], s13);
  }
}

__global__ void k_ds_terms(const int* __restrict__ ds,
                           const unsigned* __restrict__ inside,
                           const float* __restrict__ gd,
                           const float* __restrict__ v2v_min,
                           float* __restrict__ acc) {
  int j = blockIdx.x * blockDim.x + threadIdx.x;
  float s = 0.f, c = 0.f;
  if (j < NDSC) {
    int d = ds[j];
    if (!inside[d]) {
      float w = 1.0f / (5.0f * gd[d] + 1.0f);
      s = 0.04f * w * tanhf(v2v_min[d] * (1.0f / 0.04f));  // A1,A2
      c = 1.0f;
    }
  }
  float ss = wave_sum_half(s), sc = wave_sum_half(c);
  if ((threadIdx.x & 15) == 0) { atomicAdd(&acc[0], ss); atomicAdd(&acc[1], sc); }
}

__global__ void k_hand_terms(const int* __restrict__ hcp,
                             const float* __restrict__ hcpw,
                             const unsigned* __restrict__ inside,
                             const float* __restrict__ v2v_min,
                             float* __restrict__ acc) {
  int k = blockIdx.x * blockDim.x + threadIdx.x;
  float los = 0.f, loc = 0.f, ros = 0.f, roc = 0.f;
  float lis = 0.f, lic = 0.f, ris = 0.f, ric = 0.f;
  if (k < HAC) {
    int li = hcp[k];
    float vm = v2v_min[li];
    float lout = 0.01f * tanhf(vm * (1.0f / 0.01f));        // C1,C2
    float lin  = 0.023f * tanhf(vm * (1.0f / 0.02f));       // D1,D2
    float lw = 1.0f - 0.1f * hcpw[k];
    if (inside[li]) { lis = lin; lic = 1.0f; } else { los = lw * lout; loc = 1.0f; }
    int ri = hcp[k + HAC];
    float vmr = v2v_min[ri];
    float rout = 0.01f * tanhf(vmr * (1.0f / 0.01f));
    float rin  = 0.023f * tanhf(vmr * (1.0f / 0.02f));
    float rw = 1.0f - 0.1f * hcpw[k + HAC];
    if (inside[ri]) { ris = rin; ric = 1.0f; } else { ros = rw * rout; roc = 1.0f; }
  }
  float a4 = wave_sum_half(los), a5 = wave_sum_half(loc);
  float a6 = wave_sum_half(ros), a7 = wave_sum_half(roc);
  float a8 = wave_sum_half(lis), a9 = wave_sum_half(lic);
  float a10 = wave_sum_half(ris), a11 = wave_sum_half(ric);
  if ((threadIdx.x & 15) == 0) {
    atomicAdd(&acc[4], a4);  atomicAdd(&acc[5], a5);
    atomicAdd(&acc[6], a6);  atomicAdd(&acc[7], a7);
    atomicAdd(&acc[8], a8);  atomicAdd(&acc[9], a9);
    atomicAdd(&acc[10], a10); atomicAdd(&acc[11], a11);
  }
}

__global__ void k_pose_terms(const float* __restrict__ bp, const float* __restrict__ ip,
                             const float* __restrict__ lh, const float* __restrict__ rh,
                             float* __restrict__ acc) {
  int i = threadIdx.x;  // 64 threads, 2 full waves
  float p = 0.f, h = 0.f;
  if (i < 63) { float d = bp[i] - ip[i]; p = d * d; }
  if (i < 45) { h = lh[i] * lh[i] + rh[i] * rh[i]; }
  float sp = wave_sum_half(p), sh = wave_sum_half(h);
  if ((threadIdx.x & 15) == 0) { atomicAdd(&acc[14], sp); atomicAdd(&acc[15], sh); }
}

__global__ void k_finalize(const float* __restrict__ acc, float* __restrict__ out) {
  if (threadIdx.x == 0 && blockIdx.x == 0) {
    auto mm = [](float s, float c) { return c > 0.0f ? s / c : 0.0f; };
    float contactloss = 0.5f * mm(acc[0], acc[1]);            // CONTACT_W
    float insideloss  = 0.5f * mm(acc[2], acc[3]);            // INSIDE_W
    float hand = (mm(acc[8], acc[9]) + mm(acc[10], acc[11]))  // hc_in
               + (mm(acc[4], acc[5]) + mm(acc[6], acc[7]));   // hc_out, HCP_W=1
    float angle = 0.1f * mm(acc[12], acc[13]);                // ANGLE_W
    float pose = 0.01f * acc[14];                             // POSE_W
    float handpose = 0.01f * acc[15];                         // HANDPOSE_W
    float outside = 2.0f * acc[16];                           // OUTSIDE_W
    out[0] = contactloss + insideloss + outside + pose + handpose + angle + hand;
  }
}

extern "C" void kernel_launch(void* const* d_in, const int* in_sizes, int n_in,
                              void* d_out, int out_size, void* d_ws, size_t ws_size,
                              hipStream_t stream) {
  const float* vertices   = (const float*)d_in[0];
  const float* v2v_min    = (const float*)d_in[1];
  const float* geodist    = (const float*)d_in[2];
  const float* hcpw       = (const float*)d_in[3];
  const float* body_pose  = (const float*)d_in[4];
  const float* init_pose  = (const float*)d_in[5];
  const float* lh         = (const float*)d_in[6];
  const float* rh         = (const float*)d_in[7];
  const float* init_verts = (const float*)d_in[8];
  const int*   v2v_idx    = (const int*)d_in[9];
  const unsigned char* exterior = (const unsigned char*)d_in[10];  // jnp bool = 1 byte
  const int*   ds         = (const int*)d_in[11];
  const int*   contact    = (const int*)d_in[12];
  const int*   hcp        = (const int*)d_in[13];
  const int*   faces      = (const int*)d_in[14];

  char* ws = (char*)d_ws;
  float*    acc     = (float*)ws;                              // 32 floats
  float*    gd      = (float*)(ws + 128);                      // NVERT floats
  float*    vn      = (float*)(ws + 128 + 4 * NVERT);          // 3*NVERT floats
  unsigned* inside  = (unsigned*)(ws + 128 + 16 * NVERT);      // NVERT u32
  unsigned* colmask = (unsigned*)(ws + 128 + 20 * NVERT);      // COLW u32
  const int total_words = (128 + 20 * NVERT + 4 * COLW) / 4;

  k_zero<<<(total_words + 255) / 256, 256, 0, stream>>>((unsigned*)ws, total_words);
  k_scatter<<<(NDSC + 255) / 256, 256, 0, stream>>>(ds, exterior, contact, inside, colmask);
  k_gather_min<<<(NVERT + 7) / 8, 256, 0, stream>>>(geodist, colmask, gd);
  k_face_normals<<<(NFACE + 255) / 256, 256, 0, stream>>>(vertices, faces, vn);
  k_normalize_vn<<<(NVERT + 255) / 256, 256, 0, stream>>>(vn, acc);
  k_vertex_terms<<<(NVERT + 255) / 256, 256, 0, stream>>>(
      vertices, init_verts, v2v_min, v2v_idx, gd, inside, vn, acc);
  k_ds_terms<<<(NDSC + 255) / 256, 256, 0, stream>>>(ds, inside, gd, v2v_min, acc);
  k_hand_terms<<<(HAC + 255) / 256, 256, 0, stream>>>(hcp, hcpw, inside, v2v_min, acc);
  k_pose_terms<<<1, 64, 0, stream>>>(body_pose, init_pose, lh, rh, acc);
  k_finalize<<<1, 32, 0, stream>>>(acc, (float*)d_out);
}